// HyperGraphConv_55697135895082
// MI455X (gfx1250) — compile-verified
//
#include <hip/hip_runtime.h>

#define N_NODES 100000
#define N_EDGES 1600000
#define DIM     128
#define LAYERS  3
#define EPS_N   1e-12f

typedef float v2f __attribute__((ext_vector_type(2)));
typedef float v8f __attribute__((ext_vector_type(8)));

// ---------------------------------------------------------------------------
// Kernel 1: out = l2_normalize(emb); x = emb.   One wave (32 lanes) per row,
// each lane owns a float4 (32*4 = 128 = DIM).
// ---------------------------------------------------------------------------
__global__ __launch_bounds__(256) void hg_init_kernel(
    const float* __restrict__ emb, float* __restrict__ out, float* __restrict__ x)
{
    int row  = blockIdx.x * (blockDim.x >> 5) + (threadIdx.x >> 5);
    int lane = threadIdx.x & 31;
    if (row >= N_NODES) return;

    const float4 v = ((const float4*)(emb + (size_t)row * DIM))[lane];
    float ss = v.x * v.x + v.y * v.y + v.z * v.z + v.w * v.w;
    #pragma unroll
    for (int o = 16; o > 0; o >>= 1) ss += __shfl_xor(ss, o, 32);
    float inv = 1.0f / fmaxf(sqrtf(ss), EPS_N);

    float4 o4;
    o4.x = v.x * inv; o4.y = v.y * inv; o4.z = v.z * inv; o4.w = v.w * inv;
    ((float4*)(out + (size_t)row * DIM))[lane] = o4;
    ((float4*)(x   + (size_t)row * DIM))[lane] = v;
}

// ---------------------------------------------------------------------------
// Kernel 2: Y = X @ W^T using V_WMMA_F32_16X16X4_F32 (full fp32 precision).
//   Y[m][n] = sum_k X[m][k] * W[n][k]
// Block = 256 threads = 8 waves. One block per 16-row tile of X (6250 blocks).
// Wave w computes the 16x16 output tile at columns n0 = 16*w.
// W (64KB) and the 16x128 X tile (8KB) are staged in LDS.
//
// A-frag (16x4 f32, ISA layout): lane l<16 holds A[M=l][K=k0+0 / k0+1] in
// v2f; lane l+16 holds A[M=l][K=k0+2 / k0+3].  B-frag (4x16) mirrors this
// with N across lanes: lane l<16 holds B[k0+0][n0+l], B[k0+1][n0+l]; lane
// l+16 holds K=k0+2/k0+3.  Since B[k][n] = W[n][k], both frags are float2
// reads at (row-of-LDS, k0 + 2*(lane>>4)).
// ---------------------------------------------------------------------------
__global__ __launch_bounds__(256) void hg_gemm_wmma_kernel(
    const float* __restrict__ X, const float* __restrict__ W, float* __restrict__ Y)
{
    __shared__ float sW[DIM * DIM];   // W[n][k] row-major, 64 KB
    __shared__ float sX[16 * DIM];    // 16-row tile of X, 8 KB

    const int tid  = threadIdx.x;
    const int row0 = blockIdx.x * 16;

    // Stage W: 16384 floats / 256 threads = 16 float4 each
    {
        const float4* w4 = (const float4*)W;
        float4*       s4 = (float4*)sW;
        #pragma unroll
        for (int i = tid; i < (DIM * DIM) / 4; i += 256) s4[i] = w4[i];
    }
    // Stage X tile: 2048 floats = 512 float4
    {
        const float4* x4 = (const float4*)(X + (size_t)row0 * DIM);
        float4*       s4 = (float4*)sX;
        #pragma unroll
        for (int i = tid; i < (16 * DIM) / 4; i += 256) s4[i] = x4[i];
    }
    __syncthreads();

    const int wave  = tid >> 5;          // 0..7 -> column tile
    const int lane  = tid & 31;
    const int n0    = wave * 16;
    const int l15   = lane & 15;
    const int khalf = (lane >> 4) * 2;   // 0 or 2

    const float* aBase = sX + l15 * DIM + khalf;         // X[row0+l15][khalf + k0 ...]
    const float* bBase = sW + (n0 + l15) * DIM + khalf;  // W[n0+l15][khalf + k0 ...]

    v8f c = {};
    #pragma unroll
    for (int k0 = 0; k0 < DIM; k0 += 4) {
        v2f a = *(const v2f*)(aBase + k0);
        v2f b = *(const v2f*)(bBase + k0);
        c = __builtin_amdgcn_wmma_f32_16x16x4_f32(
                /*neg_a=*/false, a, /*neg_b=*/false, b,
                /*c_mod=*/(short)0, c, /*reuse_a=*/false, /*reuse_b=*/false);
    }

    // D layout: VGPR r, lanes 0-15 -> (M=r, N=lane); lanes 16-31 -> (M=r+8, N=lane-16)
    const int mhi = (lane >> 4) * 8;
    float* yb = Y + (size_t)(row0 + mhi) * DIM + n0 + l15;
    #pragma unroll
    for (int r = 0; r < 8; ++r) yb[(size_t)r * DIM] = c[r];
}

// ---------------------------------------------------------------------------
// Kernel 3: COO scatter: acc[rows[e]] += vals[e] * Y[cols[e]].
// One wave per edge; lane owns a float4 slice; hardware f32 atomics.
// ---------------------------------------------------------------------------
__global__ __launch_bounds__(256) void hg_scatter_kernel(
    const int* __restrict__ rows, const int* __restrict__ cols,
    const float* __restrict__ vals, const float* __restrict__ Y,
    float* __restrict__ acc)
{
    int e = blockIdx.x * (blockDim.x >> 5) + (threadIdx.x >> 5);
    if (e >= N_EDGES) return;
    int lane = threadIdx.x & 31;

    const int   r = rows[e];
    const int   c = cols[e];
    const float v = vals[e];

    const float4 y = ((const float4*)(Y + (size_t)c * DIM))[lane];
    float* a = acc + (size_t)r * DIM + lane * 4;
    unsafeAtomicAdd(a + 0, v * y.x);
    unsafeAtomicAdd(a + 1, v * y.y);
    unsafeAtomicAdd(a + 2, v * y.z);
    unsafeAtomicAdd(a + 3, v * y.w);
}

// ---------------------------------------------------------------------------
// Kernel 4: x = relu(acc); out = (out + l2_normalize(x)) * scale.
// scale = 1.0 for layers 0,1 and 0.25 for the last layer (folds /(L+1)).
// ---------------------------------------------------------------------------
__global__ __launch_bounds__(256) void hg_finalize_kernel(
    const float* __restrict__ acc, float* __restrict__ x,
    float* __restrict__ out, float scale)
{
    int row  = blockIdx.x * (blockDim.x >> 5) + (threadIdx.x >> 5);
    int lane = threadIdx.x & 31;
    if (row >= N_NODES) return;

    float4 v = ((const float4*)(acc + (size_t)row * DIM))[lane];
    v.x = fmaxf(v.x, 0.0f); v.y = fmaxf(v.y, 0.0f);
    v.z = fmaxf(v.z, 0.0f); v.w = fmaxf(v.w, 0.0f);
    ((float4*)(x + (size_t)row * DIM))[lane] = v;

    float ss = v.x * v.x + v.y * v.y + v.z * v.z + v.w * v.w;
    #pragma unroll
    for (int o = 16; o > 0; o >>= 1) ss += __shfl_xor(ss, o, 32);
    float inv = 1.0f / fmaxf(sqrtf(ss), EPS_N);

    float4 o4 = ((const float4*)(out + (size_t)row * DIM))[lane];
    o4.x = (o4.x + v.x * inv) * scale;
    o4.y = (o4.y + v.y * inv) * scale;
    o4.z = (o4.z + v.z * inv) * scale;
    o4.w = (o4.w + v.w * inv) * scale;
    ((float4*)(out + (size_t)row * DIM))[lane] = o4;
}

// ---------------------------------------------------------------------------
// Launcher
// ---------------------------------------------------------------------------
extern "C" void kernel_launch(void* const* d_in, const int* in_sizes, int n_in,
                              void* d_out, int out_size, void* d_ws, size_t ws_size,
                              hipStream_t stream) {
    const int*   rows = (const int*)d_in[0];
    const int*   cols = (const int*)d_in[1];
    const float* vals = (const float*)d_in[2];
    const float* emb  = (const float*)d_in[3];
    const float* W    = (const float*)d_in[4];
    float*       out  = (float*)d_out;

    const size_t nd = (size_t)N_NODES * DIM;
    float* x   = (float*)d_ws;   // current features
    float* y   = x + nd;         // GEMM output
    float* acc = y + nd;         // SpMM accumulator

    const int rowBlocks  = (N_NODES + 7) / 8;     // 8 waves (rows) per 256-thread block
    const int gemmBlocks = N_NODES / 16;          // 6250, exact
    const int edgeBlocks = (N_EDGES + 7) / 8;     // 8 waves (edges) per block

    hg_init_kernel<<<rowBlocks, 256, 0, stream>>>(emb, out, x);

    for (int l = 0; l < LAYERS; ++l) {
        hipMemsetAsync(acc, 0, nd * sizeof(float), stream);
        hg_gemm_wmma_kernel<<<gemmBlocks, 256, 0, stream>>>(
            x, W + (size_t)l * DIM * DIM, y);
        hg_scatter_kernel<<<edgeBlocks, 256, 0, stream>>>(rows, cols, vals, y, acc);
        const float scale = (l == LAYERS - 1) ? (1.0f / (LAYERS + 1)) : 1.0f;
        hg_finalize_kernel<<<rowBlocks, 256, 0, stream>>>(acc, x, out, scale);
    }
}